// QGIN_25649544692299
// MI455X (gfx1250) — compile-verified
//
#include <hip/hip_runtime.h>

#define BN_EPS   1e-5f
#define LRELU    0.01f

typedef __attribute__((ext_vector_type(2))) float v2f;
typedef __attribute__((ext_vector_type(8))) float v8f;

__device__ __forceinline__ void atomAddG(float* p, float v) {
  __hip_atomic_fetch_add(p, v, __ATOMIC_RELAXED, __HIP_MEMORY_SCOPE_AGENT);
}

// ------------------------------------------------------------------
// elementwise / utility kernels (float4-vectorized where possible)
// ------------------------------------------------------------------
__global__ __launch_bounds__(256) void k_zero4(float4* __restrict__ p, long n4) {
  long i = (long)blockIdx.x * blockDim.x + threadIdx.x;
  if (i < n4) p[i] = make_float4(0.f, 0.f, 0.f, 0.f);
}

__global__ __launch_bounds__(256) void k_gather_embed(
    const int* __restrict__ gt, const float* __restrict__ embed,
    float* __restrict__ h, int nrows) {
  long i = (long)blockIdx.x * blockDim.x + threadIdx.x;   // over nrows*32
  if (i >= (long)nrows * 32) return;
  int n = (int)(i >> 5), j = (int)(i & 31) * 4;
  *(float4*)(h + (size_t)n * 128 + j) =
      *(const float4*)(embed + (size_t)gt[n] * 128 + j);
}

// one wave (32 lanes) per edge: float4 gather of h[src], 4 f32 atomics into dst
__global__ __launch_bounds__(256) void k_scatter(
    const int* __restrict__ src, const int* __restrict__ dst,
    const float* __restrict__ w, const float* __restrict__ h,
    float* __restrict__ nH, float* __restrict__ nW, int nedges) {
  long tid = (long)blockIdx.x * blockDim.x + threadIdx.x;
  int e = (int)(tid >> 5);
  if (e >= nedges) return;
  int lane = (int)(tid & 31);
  int s = src[e], d = dst[e];
  const float4 hv = *(const float4*)(h + (size_t)s * 128 + lane * 4);
  float* np = nH + (size_t)d * 128 + lane * 4;
  atomAddG(np + 0, hv.x);
  atomAddG(np + 1, hv.y);
  atomAddG(np + 2, hv.z);
  atomAddG(np + 3, hv.w);
  if (lane < 3) atomAddG(nW + (size_t)d * 4 + lane, w[(size_t)e * 3 + lane]);
}

// weights transposed to [l][n][k], K-contiguous, K padded to 132 (W1) / 128 (W2)
__global__ __launch_bounds__(256) void k_prep_w1(
    const float* __restrict__ W1, float* __restrict__ W1t, int L) {
  long i = (long)blockIdx.x * blockDim.x + threadIdx.x;
  if (i >= (long)L * 128 * 132) return;
  int l = (int)(i / (128 * 132));
  int r = (int)(i - (long)l * 128 * 132);
  int n = r / 132, k = r - n * 132;
  W1t[i] = (k < 131) ? W1[(size_t)l * 131 * 128 + (size_t)k * 128 + n] : 0.f;
}

__global__ __launch_bounds__(256) void k_prep_w2(
    const float* __restrict__ W2, float* __restrict__ W2t, int L) {
  long i = (long)blockIdx.x * blockDim.x + threadIdx.x;
  if (i >= (long)L * 128 * 128) return;
  int l = (int)(i / (128 * 128));
  int r = (int)(i - (long)l * 128 * 128);
  int n = r >> 7, k = r & 127;
  W2t[i] = W2[(size_t)l * 128 * 128 + (size_t)k * 128 + n];
}

// ------------------------------------------------------------------
// Fused WMMA GEMM: C[N][128] = A' * Bt^T + bias  (+ BN column stats)
//   MODE 1: A'[n][k] = (1+eps)*h[n][k] + neighH[n][k]   (k < 128)
//                     = neighW[n][k-128]                 (k in 128..131, padded)
//   MODE 2: A'[n][k] = relu(a1[k]*Y1[n][k] + c1[k])
// Block = 128 threads (4 waves); wave = 16 rows x 128 cols,
// 8 accumulator tiles of V_WMMA_F32_16X16X4_F32; K fully unrolled so
// the 9 loads per K-step batch ahead of the 8 WMMAs.
// ------------------------------------------------------------------
template <int K, int MODE>
__global__ __launch_bounds__(128) void k_gemm(
    const float* __restrict__ P0,   // MODE1: h      | MODE2: Y1
    const float* __restrict__ P1,   // MODE1: neighH | MODE2: a1
    const float* __restrict__ P2,   // MODE1: neighW | MODE2: c1
    const float* __restrict__ epsArr, int l,
    const float* __restrict__ Bt,
    const float* __restrict__ bias,
    float* __restrict__ C,
    float* __restrict__ gsum, float* __restrict__ gsq,
    int nrows) {
  __shared__ float ssum[128];
  __shared__ float ssq[128];
  int tid = threadIdx.x;
  ssum[tid] = 0.f;
  ssq[tid] = 0.f;
  __syncthreads();

  const int lane = tid & 31;
  const int wave = tid >> 5;
  const int row0 = blockIdx.x * 64 + wave * 16;
  const int m    = lane & 15;            // A row within tile / B column
  const int kb   = (lane & 16) ? 2 : 0;  // K sub-offset (ISA 16x4 f32 layout)

  int arow = row0 + m;
  if (arow >= nrows) arow = nrows - 1;   // clamp; invalid rows masked at store

  float e1 = 1.f;
  if (MODE == 1) e1 = 1.f + epsArr[l];

  const float* ap0 = P0 + (size_t)arow * 128 + kb;
  const float* ap1 = (MODE == 1) ? (P1 + (size_t)arow * 128 + kb) : (P1 + kb);
  const float* ap2 = (MODE == 1) ? (P2 + (size_t)arow * 4 + kb) : (P2 + kb);

  const float* brow[8];
#pragma unroll
  for (int t = 0; t < 8; ++t) brow[t] = Bt + (size_t)(t * 16 + m) * K + kb;

  v8f acc[8];
#pragma unroll
  for (int t = 0; t < 8; ++t)
#pragma unroll
    for (int i = 0; i < 8; ++i) acc[t][i] = 0.f;

#pragma unroll
  for (int k0 = 0; k0 < K; k0 += 4) {
    v2f av;
    if (MODE == 1) {
      if (k0 < 128) {
        v2f hv = *(const v2f*)(ap0 + k0);
        v2f nv = *(const v2f*)(ap1 + k0);
        v2f ev = {e1, e1};
        av = ev * hv + nv;
      } else {
        av = *(const v2f*)(ap2);         // zero-padded neighW row
      }
    } else {
      v2f y  = *(const v2f*)(ap0 + k0);
      v2f aa = *(const v2f*)(ap1 + k0);
      v2f cc = *(const v2f*)(ap2 + k0);
      v2f t2 = aa * y + cc;
      av[0] = fmaxf(t2[0], 0.f);
      av[1] = fmaxf(t2[1], 0.f);
    }
    v2f bv[8];
#pragma unroll
    for (int t = 0; t < 8; ++t) bv[t] = *(const v2f*)(brow[t] + k0);
#pragma unroll
    for (int t = 0; t < 8; ++t)
      acc[t] = __builtin_amdgcn_wmma_f32_16x16x4_f32(
          false, av, false, bv[t], (short)0, acc[t], false, false);
  }

  // C/D layout: lane l, vgpr i -> row (l&16 ? 8:0)+i, col l&15
  const int mbase = (lane & 16) ? 8 : 0;
  const int nn = lane & 15;
#pragma unroll
  for (int t = 0; t < 8; ++t) {
    const int col = t * 16 + nn;
    const float bsv = bias[col];
    float s = 0.f, s2 = 0.f;
#pragma unroll
    for (int i = 0; i < 8; ++i) {
      int row = row0 + mbase + i;
      if (row < nrows) {
        float y = acc[t][i] + bsv;
        C[(size_t)row * 128 + col] = y;
        s += y;
        s2 += y * y;
      }
    }
    atomicAdd(&ssum[col], s);   // ds_add_f32
    atomicAdd(&ssq[col], s2);
  }
  __syncthreads();
  atomAddG(&gsum[tid], ssum[tid]);
  atomAddG(&gsq[tid], ssq[tid]);
}

// fold BN batch stats + gamma/beta into per-column affine a*y + c
__global__ __launch_bounds__(128) void k_bn_params(
    const float* __restrict__ gsum, const float* __restrict__ gsq,
    const float* __restrict__ gamma, const float* __restrict__ beta,
    float* __restrict__ a, float* __restrict__ c, int nrows) {
  int j = threadIdx.x;
  float inv = 1.f / (float)nrows;
  float mu = gsum[j] * inv;
  float var = gsq[j] * inv - mu * mu;
  float aj = gamma[j] * rsqrtf(var + BN_EPS);
  a[j] = aj;
  c[j] = beta[j] - aj * mu;
}

__global__ __launch_bounds__(256) void k_bn_lrelu4(
    const float* __restrict__ Y, const float* __restrict__ a,
    const float* __restrict__ c, float* __restrict__ out, long total4) {
  long i = (long)blockIdx.x * blockDim.x + threadIdx.x;
  if (i >= total4) return;
  int j = (int)(i & 31) * 4;
  float4 y = ((const float4*)Y)[i];
  float4 av = *(const float4*)(a + j);
  float4 cv = *(const float4*)(c + j);
  float4 r;
  float v;
  v = av.x * y.x + cv.x; r.x = v > 0.f ? v : LRELU * v;
  v = av.y * y.y + cv.y; r.y = v > 0.f ? v : LRELU * v;
  v = av.z * y.z + cv.z; r.z = v > 0.f ? v : LRELU * v;
  v = av.w * y.w + cv.w; r.w = v > 0.f ? v : LRELU * v;
  ((float4*)out)[i] = r;
}

// ------------------------------------------------------------------
extern "C" void kernel_launch(void* const* d_in, const int* in_sizes, int n_in,
                              void* d_out, int out_size, void* d_ws, size_t ws_size,
                              hipStream_t stream) {
  const int N = in_sizes[0];
  const int E = in_sizes[1];
  const int Lc = 3;

  const int*   gate_types = (const int*)d_in[0];
  const int*   src   = (const int*)d_in[1];
  const int*   dst   = (const int*)d_in[2];
  const float* wEdge = (const float*)d_in[3];
  const float* embed = (const float*)d_in[4];
  const float* W1    = (const float*)d_in[5];
  const float* b1    = (const float*)d_in[6];
  const float* g1    = (const float*)d_in[7];
  const float* be1   = (const float*)d_in[8];
  const float* W2    = (const float*)d_in[9];
  const float* b2    = (const float*)d_in[10];
  const float* g2    = (const float*)d_in[11];
  const float* be2   = (const float*)d_in[12];
  const float* epsA  = (const float*)d_in[13];
  float* out = (float*)d_out;

  auto al = [](size_t x) { return (x + 255) & ~(size_t)255; };
  char* base = (char*)d_ws;
  size_t off = 0;
  float* neighH = (float*)(base + off); off += al((size_t)N * 128 * 4);  // later Y2
  float* neighW = (float*)(base + off); off += al((size_t)N * 4 * 4);
  float* Y1     = (float*)(base + off); off += al((size_t)N * 128 * 4);
  float* W1t    = (float*)(base + off); off += al((size_t)Lc * 128 * 132 * 4);
  float* W2t    = (float*)(base + off); off += al((size_t)Lc * 128 * 128 * 4);
  float* stats  = (float*)(base + off); off += al((size_t)8 * 128 * 4);
  (void)ws_size; (void)n_in; (void)out_size;

  float* Y2   = neighH;
  float* sum1 = stats;        float* sq1 = stats + 128;
  float* sum2 = stats + 256;  float* sq2 = stats + 384;
  float* a1   = stats + 512;  float* c1  = stats + 640;
  float* a2   = stats + 768;  float* c2  = stats + 896;

  auto cdiv = [](long a, long b) { return (int)((a + b - 1) / b); };

  k_gather_embed<<<cdiv((long)N * 32, 256), 256, 0, stream>>>(gate_types, embed, out, N);
  k_prep_w1<<<cdiv((long)Lc * 128 * 132, 256), 256, 0, stream>>>(W1, W1t, Lc);
  k_prep_w2<<<cdiv((long)Lc * 128 * 128, 256), 256, 0, stream>>>(W2, W2t, Lc);

  const int gemmBlocks = cdiv(N, 64);
  for (int l = 0; l < Lc; ++l) {
    k_zero4<<<cdiv((long)N * 32, 256), 256, 0, stream>>>((float4*)neighH, (long)N * 32);
    k_zero4<<<cdiv((long)N, 256), 256, 0, stream>>>((float4*)neighW, (long)N);
    k_zero4<<<1, 128, 0, stream>>>((float4*)stats, 128);

    k_scatter<<<cdiv((long)E * 32, 256), 256, 0, stream>>>(src, dst, wEdge, out,
                                                           neighH, neighW, E);
    k_gemm<132, 1><<<gemmBlocks, 128, 0, stream>>>(
        out, neighH, neighW, epsA, l, W1t + (size_t)l * 128 * 132,
        b1 + l * 128, Y1, sum1, sq1, N);
    k_bn_params<<<1, 128, 0, stream>>>(sum1, sq1, g1 + l * 128, be1 + l * 128, a1, c1, N);
    k_gemm<128, 2><<<gemmBlocks, 128, 0, stream>>>(
        Y1, a1, c1, epsA, l, W2t + (size_t)l * 128 * 128,
        b2 + l * 128, Y2, sum2, sq2, N);
    k_bn_params<<<1, 128, 0, stream>>>(sum2, sq2, g2 + l * 128, be2 + l * 128, a2, c2, N);
    k_bn_lrelu4<<<cdiv((long)N * 32, 256), 256, 0, stream>>>(Y2, a2, c2, out, (long)N * 32);
  }
}